// GraphNetwork_1769526526151
// MI455X (gfx1250) — compile-verified
//
#include <hip/hip_runtime.h>
#include <math.h>

#define Nn 20000
#define Ee 640000
#define Hh 256
#define Rr 10
#define Cc 7

typedef float v2f __attribute__((ext_vector_type(2)));
typedef float v8f __attribute__((ext_vector_type(8)));

// ---------------------------------------------------------------- utilities
__global__ __launch_bounds__(256) void k_zero_f(float* __restrict__ p, int n) {
  int i = blockIdx.x * 256 + threadIdx.x;
  if (i < n) p[i] = 0.0f;
}
__global__ __launch_bounds__(256) void k_zero_i(int* __restrict__ p, int n) {
  int i = blockIdx.x * 256 + threadIdx.x;
  if (i < n) p[i] = 0;
}

// ------------------------------------------------------------- CSR building
__global__ __launch_bounds__(256) void k_hist(const int* __restrict__ dsti,
                                              int* __restrict__ deg) {
  int e = blockIdx.x * 256 + threadIdx.x;
  if (e < Ee) atomicAdd(&deg[dsti[e]], 1);
}

// single-block exclusive scan over N node degrees -> rowstart[N+1], cursor[N]
__global__ __launch_bounds__(1024) void k_scan(const int* __restrict__ deg,
                                               int* __restrict__ rowstart,
                                               int* __restrict__ cursor) {
  __shared__ int part[1024];
  const int t = threadIdx.x;
  const int chunk = (Nn + 1023) / 1024;           // 20
  int b = t * chunk;
  int e = b + chunk; if (e > Nn) e = Nn;
  int s = 0;
  for (int i = b; i < e; ++i) s += deg[i];
  part[t] = s;
  __syncthreads();
  // Hillis-Steele inclusive scan over 1024 partials
  for (int off = 1; off < 1024; off <<= 1) {
    int v = part[t];
    int u = (t >= off) ? part[t - off] : 0;
    __syncthreads();
    part[t] = v + u;
    __syncthreads();
  }
  int run = (t == 0) ? 0 : part[t - 1];
  for (int i = b; i < e; ++i) {
    rowstart[i] = run; cursor[i] = run;
    run += deg[i];
  }
  if (t == 1023) rowstart[Nn] = part[1023];       // == Ee
}

__global__ __launch_bounds__(256) void k_scatter(const int* __restrict__ dsti,
                                                 int* __restrict__ cursor,
                                                 int* __restrict__ eidx) {
  int e = blockIdx.x * 256 + threadIdx.x;
  if (e >= Ee) return;
  int pos = atomicAdd(&cursor[dsti[e]], 1);
  eidx[pos] = e;
}

// ------------------------------------------------------------- fp32 WMMA GEMM
// C[M x 256] = act( A[M x K](lda) * B[K x 256] + bias )
// One wave computes an 80x16 strip: 5 row-tiles share each B fragment.
// Requires M % 80 == 0 (20000 = 250*80).
__global__ __launch_bounds__(256)
void k_gemm(const float* __restrict__ A, int lda, int M,
            const float* __restrict__ B, int K,
            const float* __restrict__ bias, int relu,
            float* __restrict__ C, int ldc) {
  const int MB = 5;
  int wave = (blockIdx.x * 256 + threadIdx.x) >> 5;
  int lane = threadIdx.x & 31;
  int mw = wave >> 4;                 // 80-row strip index (16 col-tiles)
  int ct = wave & 15;                 // column tile
  int row0 = mw * (16 * MB);
  int col0 = ct * 16;
  if (row0 >= M) return;

  const int lhalf = lane & 15;
  const int khalf = (lane >> 4) << 1; // 0 for lanes 0-15, 2 for lanes 16-31

  v8f acc[MB];
#pragma unroll
  for (int i = 0; i < MB; ++i) { v8f z = {}; acc[i] = z; }

  const float* bcol = B + col0 + lhalf;
  for (int k = 0; k < K; k += 4) {
    // B 4x16 fragment: VGPR v holds row K = k + v (+2 in upper half-wave)
    v2f bfrag;
    bfrag.x = bcol[(size_t)(k + khalf) * 256];
    bfrag.y = bcol[(size_t)(k + khalf + 1) * 256];
#pragma unroll
    for (int i = 0; i < MB; ++i) {
      // A 16x4 fragment: lane -> row (M = lane&15), v0/v1 = K pair
      const float* ap = A + (size_t)(row0 + i * 16 + lhalf) * lda + k + khalf;
      v2f afrag; afrag.x = ap[0]; afrag.y = ap[1];
      acc[i] = __builtin_amdgcn_wmma_f32_16x16x4_f32(
          false, afrag, false, bfrag, (short)0, acc[i], false, false);
    }
  }

  float bv = bias ? bias[col0 + lhalf] : 0.0f;
  const int rhalf = (lane >> 4) << 3;  // +8 rows for upper half-wave
#pragma unroll
  for (int i = 0; i < MB; ++i) {
#pragma unroll
    for (int v = 0; v < 8; ++v) {
      int row = row0 + i * 16 + rhalf + v;
      float val = acc[i][v] + bv;
      if (relu) val = fmaxf(val, 0.0f);
      C[(size_t)row * ldc + col0 + lhalf] = val;
    }
  }
}

// --------------------------------------------- per-relation scatter-sum (CSR)
// One wave per destination node; exclusive non-atomic accumulate.
__global__ __launch_bounds__(256)
void k_aggr(const float* __restrict__ xw,        // [N,256] for relation rel
            const int* __restrict__ rowstart,    // [N+1]
            const int* __restrict__ eidx,        // [E] edge ids grouped by dst
            const int* __restrict__ srcj,        // edge_index[1]
            const int* __restrict__ etype,
            const float* __restrict__ enorm,
            int rel, float* __restrict__ accum) {
  int node = (blockIdx.x * 256 + threadIdx.x) >> 5;
  if (node >= Nn) return;
  int lane = threadIdx.x & 31;
  int s = rowstart[node], t = rowstart[node + 1];

  float4 s0 = {0, 0, 0, 0}, s1 = {0, 0, 0, 0};
  int hits = 0;
  for (int p = s; p < t; ++p) {           // wave-uniform loop
    int eid = eidx[p];
    if (etype[eid] != rel) continue;
    ++hits;
    float nw = enorm[eid] * 2.0f;         // forward passes edge_norm * 2
    const float* row = xw + (size_t)srcj[eid] * Hh + lane * 8;
    float4 a = *(const float4*)row;
    float4 b = *(const float4*)(row + 4);
    s0.x += nw * a.x; s0.y += nw * a.y; s0.z += nw * a.z; s0.w += nw * a.w;
    s1.x += nw * b.x; s1.y += nw * b.y; s1.z += nw * b.z; s1.w += nw * b.w;
  }
  if (hits) {
    float* d = accum + (size_t)node * Hh + lane * 8;
    float4 d0 = *(float4*)d;
    float4 d1 = *(float4*)(d + 4);
    d0.x += s0.x; d0.y += s0.y; d0.z += s0.z; d0.w += s0.w;
    d1.x += s1.x; d1.y += s1.y; d1.z += s1.z; d1.w += s1.w;
    *(float4*)d = d0;
    *(float4*)(d + 4) = d1;
  }
}

// ----------------------------------------------------------------- epilogues
__global__ __launch_bounds__(256) void k_finish1(float* __restrict__ accum,
                                                 const int* __restrict__ deg) {
  int idx = blockIdx.x * 256 + threadIdx.x;
  if (idx >= Nn * Hh) return;
  float inv = 1.0f / fmaxf((float)deg[idx >> 8], 1.0f);
  float v = accum[idx] * inv;
  accum[idx] = 1.0f / (1.0f + expf(-v));          // out1 in place
}

__global__ __launch_bounds__(256) void k_finish2(const float* __restrict__ accum,
                                                 const int* __restrict__ deg,
                                                 float* __restrict__ em) {
  int idx = blockIdx.x * 256 + threadIdx.x;
  if (idx >= Nn * Hh) return;
  int i = idx >> 8, h = idx & 255;
  float inv = 1.0f / fmaxf((float)deg[i], 1.0f);
  em[(size_t)i * (2 * Hh) + Hh + h] = accum[idx] * inv;   // out2 half
}

__global__ __launch_bounds__(256) void k_copyx(const float* __restrict__ x,
                                               float* __restrict__ xsec,
                                               float* __restrict__ em) {
  int idx = blockIdx.x * 256 + threadIdx.x;
  if (idx >= Nn * Hh) return;
  float v = x[idx];
  xsec[idx] = v;
  em[(size_t)(idx >> 8) * (2 * Hh) + (idx & 255)] = v;    // x half of emotions
}

__global__ __launch_bounds__(256)
void k_fc_logsoftmax(const float* __restrict__ hidden,
                     const float* __restrict__ fcW,   // [256,7]
                     const float* __restrict__ fcb,
                     float* __restrict__ out) {       // [N,7]
  int n = blockIdx.x * 256 + threadIdx.x;
  if (n >= Nn) return;
  float l[Cc];
#pragma unroll
  for (int c = 0; c < Cc; ++c) l[c] = fcb[c];
  const float* h = hidden + (size_t)n * Hh;
  for (int d = 0; d < Hh; ++d) {
    float hv = h[d];
    const float* wr = fcW + d * Cc;                 // lane-uniform, cache hot
#pragma unroll
    for (int c = 0; c < Cc; ++c) l[c] += hv * wr[c];
  }
  float m = l[0];
#pragma unroll
  for (int c = 1; c < Cc; ++c) m = fmaxf(m, l[c]);
  float sum = 0.0f;
#pragma unroll
  for (int c = 0; c < Cc; ++c) sum += expf(l[c] - m);
  float lse = m + logf(sum);
#pragma unroll
  for (int c = 0; c < Cc; ++c) out[(size_t)n * Cc + c] = l[c] - lse;
}

// ------------------------------------------------------------------- driver
extern "C" void kernel_launch(void* const* d_in, const int* in_sizes, int n_in,
                              void* d_out, int out_size, void* d_ws, size_t ws_size,
                              hipStream_t stream) {
  (void)in_sizes; (void)n_in; (void)out_size; (void)ws_size;

  const float* x       = (const float*)d_in[1];
  const int*   eindex  = (const int*)d_in[2];      // [2,E]
  const float* enorm   = (const float*)d_in[3];
  const int*   etype   = (const int*)d_in[4];
  const float* w1      = (const float*)d_in[7];    // [R,H,H]
  const float* w2      = (const float*)d_in[8];
  const float* linW    = (const float*)d_in[9];    // [2H,H]
  const float* linb    = (const float*)d_in[10];
  const float* fcW     = (const float*)d_in[11];   // [H,C]
  const float* fcb     = (const float*)d_in[12];

  const int* dsti = eindex;         // i = edge_index[0]
  const int* srcj = eindex + Ee;    // j = edge_index[1]

  // workspace layout (floats first, then ints) — ~64 MB
  float* xw      = (float*)d_ws;                // N*H  (also reused as hidden)
  float* accum1  = xw + (size_t)Nn * Hh;        // N*H  (becomes out1 in place)
  float* accum2  = accum1 + (size_t)Nn * Hh;    // N*H
  int*   deg     = (int*)(accum2 + (size_t)Nn * Hh);  // N
  int*   rowstart= deg + Nn;                    // N+1
  int*   cursor  = rowstart + Nn + 1;           // N
  int*   eidx    = cursor + Nn;                 // E

  float* outLog  = (float*)d_out;                         // [N,7]
  float* outX    = outLog + (size_t)Nn * Cc;              // [N,256]
  float* outEm   = outX + (size_t)Nn * Hh;                // [N,512]

  const int gNH = (Nn * Hh + 255) / 256;   // 20000
  const int gE  = (Ee + 255) / 256;        // 2500
  const int gN  = (Nn + 255) / 256;        // 79
  const int gWaveNode = (Nn * 32 + 255) / 256;            // 2500 (wave/node)
  const int gGemm = ((Nn / 80) * 16) / 8;                 // 500 blocks

  // --- init + CSR build (once per call; shared by both layers) ---
  k_zero_f<<<gNH, 256, 0, stream>>>(accum1, Nn * Hh);
  k_zero_f<<<gNH, 256, 0, stream>>>(accum2, Nn * Hh);
  k_zero_i<<<gN, 256, 0, stream>>>(deg, Nn);
  k_hist<<<gE, 256, 0, stream>>>(dsti, deg);
  k_scan<<<1, 1024, 0, stream>>>(deg, rowstart, cursor);
  k_scatter<<<gE, 256, 0, stream>>>(dsti, cursor, eidx);

  // x -> output sections (x copy + left half of emotions)
  k_copyx<<<gNH, 256, 0, stream>>>(x, outX, outEm);

  // --- layer 1: stream relations; xw_r stays L2-resident for the gather ---
  for (int r = 0; r < Rr; ++r) {
    k_gemm<<<gGemm, 256, 0, stream>>>(x, Hh, Nn, w1 + (size_t)r * Hh * Hh, Hh,
                                      nullptr, 0, xw, Hh);
    k_aggr<<<gWaveNode, 256, 0, stream>>>(xw, rowstart, eidx, srcj, etype,
                                          enorm, r, accum1);
  }
  k_finish1<<<gNH, 256, 0, stream>>>(accum1, deg);   // out1 = sigmoid(mean)

  // --- layer 2 ---
  for (int r = 0; r < Rr; ++r) {
    k_gemm<<<gGemm, 256, 0, stream>>>(accum1, Hh, Nn, w2 + (size_t)r * Hh * Hh,
                                      Hh, nullptr, 0, xw, Hh);
    k_aggr<<<gWaveNode, 256, 0, stream>>>(xw, rowstart, eidx, srcj, etype,
                                          enorm, r, accum2);
  }
  k_finish2<<<gNH, 256, 0, stream>>>(accum2, deg, outEm);  // out2 -> emotions

  // --- hidden = relu(emotions @ lin_W + b) ; reuse xw as hidden ---
  k_gemm<<<gGemm, 256, 0, stream>>>(outEm, 2 * Hh, Nn, linW, 2 * Hh,
                                    linb, 1, xw, Hh);

  // --- logits + log_softmax ---
  k_fc_logsoftmax<<<gN, 256, 0, stream>>>(xw, fcW, fcb, outLog);
}